// CompletePatchReadout_44968307589151
// MI455X (gfx1250) — compile-verified
//
#include <hip/hip_runtime.h>
#include <stdint.h>

// Problem constants (CompletePatchReadout)
#define BATCH   32
#define TSTEPS  12
#define NPATCH  128
#define NHID    128
#define KDIM    (TSTEPS * NHID)        // 1536
#define NODESP  64
#define HORIZON 12
#define NOUT    (NODESP * HORIZON)     // 768 per patch
#define NNODES  (NPATCH * NODESP)      // 8192

// Tiling
#define NB      64                     // output cols per workgroup
#define KB      128                    // K chunk (= one timestep of x)
#define NCHUNKS (KDIM / KB)            // 12
#define XS_LD   132                    // padded LDS stride for X (conflict-free b64 A reads)
#define WS_LD   72                     // padded LDS stride for W (2*72 % 64 == 16 -> disjoint bank halves)

#define XBUF_F  (BATCH * XS_LD)        // 4224 floats per X buffer
#define WBUF_F  (KB * WS_LD)           // 9216 floats per W buffer
#define SMEM_F  (2 * XBUF_F + 2 * WBUF_F)   // 26880 floats = 107520 bytes

typedef float v2f __attribute__((ext_vector_type(2)));
typedef float v8f __attribute__((ext_vector_type(8)));

// Async copy global -> LDS, 16 bytes per lane, GVS addressing (SGPR base + VGPR offset).
// Tracked with ASYNCcnt; LDS dest offset (bytes from workgroup LDS base) rides in the VDST VGPR.
// RT variant: X tiles (reused by 12 sibling workgroups via L2 -> keep temporal).
__device__ __forceinline__ void async_load_b128_rt(unsigned ldsByteOff, const float* base, unsigned gByteOff) {
    asm volatile("global_load_async_to_lds_b128 %0, %1, %2"
                 :
                 : "v"(ldsByteOff), "v"(gByteOff), "s"(base)
                 : "memory");
}

// NT variant: W is a 603 MB read-once stream -> non-temporal so it doesn't evict reusable X lines.
__device__ __forceinline__ void async_load_b128_nt(unsigned ldsByteOff, const float* base, unsigned gByteOff) {
    asm volatile("global_load_async_to_lds_b128 %0, %1, %2 th:TH_LOAD_NT"
                 :
                 : "v"(ldsByteOff), "v"(gByteOff), "s"(base)
                 : "memory");
}

__device__ __forceinline__ void issue_chunk(int kc, int buf, int tid,
                                            const float* Xp, const float* Wp) {
    // ---- X chunk: 32 rows x 128 contiguous floats (timestep kc) = 1024 b128 transfers ----
    #pragma unroll
    for (int i = 0; i < 4; ++i) {
        const int idx = tid + i * 256;
        const int row = idx >> 5;         // batch 0..31
        const int c4  = idx & 31;         // float4 col
        const unsigned goff = (unsigned)((((row * TSTEPS + kc) * (NPATCH * NHID)) + c4 * 4) * 4);
        const unsigned loff = (unsigned)((buf * XBUF_F + row * XS_LD + c4 * 4) * 4);
        async_load_b128_rt(loff, Xp, goff);
    }
    // ---- W chunk: 128 rows x 64 floats = 2048 b128 transfers, coalesced rows, non-temporal ----
    #pragma unroll
    for (int i = 0; i < 8; ++i) {
        const int idx = tid + i * 256;
        const int row = idx >> 4;         // k within chunk
        const int c4  = idx & 15;
        const unsigned goff = (unsigned)(((((kc * KB + row) * NOUT)) + c4 * 4) * 4);
        const unsigned loff = (unsigned)((2 * XBUF_F + buf * WBUF_F + row * WS_LD + c4 * 4) * 4);
        async_load_b128_nt(loff, Wp, goff);
    }
}

__global__ __launch_bounds__(256)
void CompletePatchReadout_44968307589151_kernel(
    const float* __restrict__ x,        // [B, T, P, F]
    const float* __restrict__ W,        // [P, K, NOUT]
    const float* __restrict__ bias,     // [P, NOUT]
    const long long* __restrict__ map,  // [P, NODESP] int64
    float* __restrict__ out)            // [B, NNODES, HORIZON]
{
    extern __shared__ __align__(16) float smem[];   // [Xbuf0 | Xbuf1 | Wbuf0 | Wbuf1], offset 0 in LDS

    const int tid    = threadIdx.x;
    const int p      = blockIdx.x / (NOUT / NB);        // patch
    const int nchunk = blockIdx.x % (NOUT / NB);
    const int nbase  = nchunk * NB;                     // global col base (0..767)

    const int wave = tid >> 5;
    const int lane = tid & 31;
    const int mt   = wave & 1;          // M tile (batch 0..15 / 16..31)
    const int nt   = wave >> 1;         // N tile within chunk (0..3)
    const int n0   = nt * 16;

    const int half = lane >> 4;         // 0: lanes 0-15, 1: lanes 16-31
    const int koff = half * 2;          // A/B fragment K sub-offset (ISA 7.12.2)
    const int l15  = lane & 15;
    const int arow = mt * 16 + l15;     // batch row for A fragment
    const int bcol = n0 + l15;          // LDS col for B fragment

    const float* Wp = W + (size_t)p * KDIM * NOUT + nbase;  // row k: Wp[k*NOUT + n], n in [0,NB)
    const float* Xp = x + (size_t)p * NHID;                 // x[(b*T+t)*P*F + p*F + f]

    v8f acc = {};   // D = A*B + C accumulator, 16x16 f32 tile

    // Pipeline prologue: chunk 0 -> buffer 0
    issue_chunk(0, 0, tid, Xp, Wp);

    for (int kc = 0; kc < NCHUNKS; ++kc) {
        const int buf = kc & 1;
        if (kc + 1 < NCHUNKS) {
            // Overlap: issue chunk kc+1 into the other buffer (12 async ops),
            // then wait until only those 12 remain in flight (chunk kc arrived; in-order done).
            issue_chunk(kc + 1, buf ^ 1, tid, Xp, Wp);
            asm volatile("s_wait_asynccnt 0xc" ::: "memory");
        } else {
            asm volatile("s_wait_asynccnt 0x0" ::: "memory");
        }
        __syncthreads();   // every wave has waited for its own chunk-kc loads -> all data visible

        const float* Xb = smem + buf * XBUF_F;
        const float* Wb = smem + 2 * XBUF_F + buf * WBUF_F;

        // ---- 32 x v_wmma_f32_16x16x4_f32 over this K chunk ----
        #pragma unroll
        for (int kk = 0; kk < KB; kk += 4) {
            // A fragment (16x4): VGPR0 = K=koff, VGPR1 = K=koff+1 (per lane half)
            const v2f a = *(const v2f*)(&Xb[arow * XS_LD + kk + koff]);
            // B fragment (4x16): VGPR0 = row K=koff, VGPR1 = row K=koff+1
            v2f b;
            b.x = Wb[(kk + koff    ) * WS_LD + bcol];
            b.y = Wb[(kk + koff + 1) * WS_LD + bcol];
            acc = __builtin_amdgcn_wmma_f32_16x16x4_f32(
                /*neg_a=*/false, a, /*neg_b=*/false, b,
                /*c_mod=*/(short)0, acc, /*reuse_a=*/false, /*reuse_b=*/false);
        }
        __syncthreads();   // all waves done reading buf before chunk kc+2 overwrites it
    }

    // ---- epilogue: bias add + int64 permutation scatter ----
    // C/D layout: VGPR v -> M = v (lanes 0-15) or M = v+8 (lanes 16-31), N = lane%16
    const int o = nbase + n0 + l15;               // global output col 0..767
    const int j = o / HORIZON;                    // patch-local node
    const int h = o - j * HORIZON;                // horizon step
    const long long g = map[(size_t)p * NODESP + j];   // global node id
    const float bv = bias[(size_t)p * NOUT + o];
    const int mbase = mt * 16 + half * 8;

    #pragma unroll
    for (int v = 0; v < 8; ++v) {
        const int batch = mbase + v;
        out[((size_t)batch * NNODES + (size_t)g) * HORIZON + h] = acc[v] + bv;
    }
}

extern "C" void kernel_launch(void* const* d_in, const int* in_sizes, int n_in,
                              void* d_out, int out_size, void* d_ws, size_t ws_size,
                              hipStream_t stream) {
    const float*     x    = (const float*)d_in[0];
    const float*     W    = (const float*)d_in[1];
    const float*     bias = (const float*)d_in[2];
    const long long* map  = (const long long*)d_in[3];
    float*           out  = (float*)d_out;

    (void)in_sizes; (void)n_in; (void)out_size; (void)d_ws; (void)ws_size;

    dim3 grid(NPATCH * (NOUT / NB));   // 128 patches x 12 N-chunks = 1536 blocks
    dim3 block(256);                   // 8 wave32s: 2 M-tiles x 4 N-tiles
    size_t smemBytes = SMEM_F * sizeof(float);   // 107520 B: double-buffered X + W
    CompletePatchReadout_44968307589151_kernel<<<grid, block, smemBytes, stream>>>(x, W, bias, map, out);
}